// TwinTowerFull_83717502533722
// MI455X (gfx1250) — compile-verified
//
#include <hip/hip_runtime.h>
#include <math.h>

#define B_ROWS 16384
#define KHIST  50
#define DIN    384
#define KT     32
#define XS_STRIDE 33
#define WS_STRIDE 33

typedef float v2f __attribute__((ext_vector_type(2)));
typedef float v8f __attribute__((ext_vector_type(8)));

// ---------------------------------------------------------------------------
// Kernel 1: feature assembly. One block (128 threads) per batch row.
// ---------------------------------------------------------------------------
__global__ void build_features(
    const float* __restrict__ hist_num_feats, const float* __restrict__ user_stats,
    const float* __restrict__ last_content,
    const float* __restrict__ user_id_emb, const float* __restrict__ item_id_emb,
    const float* __restrict__ env_emb, const float* __restrict__ device_emb,
    const float* __restrict__ os_emb, const float* __restrict__ country_emb,
    const float* __restrict__ region_emb, const float* __restrict__ referrer_emb,
    const float* __restrict__ month_emb, const float* __restrict__ day_emb,
    const float* __restrict__ hour_emb, const float* __restrict__ minute_emb,
    const float* __restrict__ category_emb,
    const float* __restrict__ hist_num_w, const float* __restrict__ hist_num_b,
    const float* __restrict__ user_stat_w, const float* __restrict__ user_stat_b,
    const float* __restrict__ content_w, const float* __restrict__ content_b,
    const int* __restrict__ user_id, const int* __restrict__ env,
    const int* __restrict__ device_group, const int* __restrict__ os_,
    const int* __restrict__ country, const int* __restrict__ region,
    const int* __restrict__ referrer, const int* __restrict__ month,
    const int* __restrict__ day, const int* __restrict__ hour,
    const int* __restrict__ minute, const int* __restrict__ history_ids,
    const int* __restrict__ last_cat,
    float* __restrict__ x)
{
    const int b = blockIdx.x;
    const int t = threadIdx.x;

    __shared__ float s_lc[256];   // last_content row (250 used)
    __shared__ float s_w[64];     // raw exp-decay history weights (50 used)

    for (int c = t; c < 250; c += 128) s_lc[c] = last_content[(long long)b * 250 + c];
    if (t < KHIST) s_w[t] = expf(-0.5f * (float)(KHIST - 1 - t));
    __syncthreads();

    for (int d = t; d < DIN; d += 128) {
        float v;
        if (d < 64) {
            v = user_id_emb[(long long)user_id[b] * 64 + d];
        } else if (d < 160) {                      // six 16-dim categorical embs
            const int which = (d - 64) >> 4, off = (d - 64) & 15;
            const float* tab; int idx;
            switch (which) {
                case 0:  tab = env_emb;      idx = env[b];          break;
                case 1:  tab = device_emb;   idx = device_group[b]; break;
                case 2:  tab = os_emb;       idx = os_[b];          break;
                case 3:  tab = country_emb;  idx = country[b];      break;
                case 4:  tab = region_emb;   idx = region[b];       break;
                default: tab = referrer_emb; idx = referrer[b];     break;
            }
            v = tab[(long long)idx * 16 + off];
        } else if (d < 192) {                      // four 8-dim time embs
            const int which = (d - 160) >> 3, off = (d - 160) & 7;
            const float* tab; int idx;
            switch (which) {
                case 0:  tab = month_emb;  idx = month[b];  break;
                case 1:  tab = day_emb;    idx = day[b];    break;
                case 2:  tab = hour_emb;   idx = hour[b];   break;
                default: tab = minute_emb; idx = minute[b]; break;
            }
            v = tab[(long long)idx * 8 + off];
        } else if (d < 256) {                      // history pooling (64)
            const int e = d - 192;
            float wsum = 0.f;
            for (int k = 0; k < KHIST; ++k) wsum += s_w[k];
            float acc = 0.f;
            const int* hrow = history_ids + (long long)b * KHIST;
            for (int k = 0; k < KHIST; ++k) {
                const int id = hrow[k];
                acc += item_id_emb[(long long)id * 64 + e] * s_w[k];
            }
            v = acc / (wsum + 1e-9f);
        } else if (d < 272) {                      // hist_num proj (16)
            const int j = d - 256;
            v = hist_num_b[j] + hist_num_w[j * 2] * hist_num_feats[(long long)b * 2]
                              + hist_num_w[j * 2 + 1] * hist_num_feats[(long long)b * 2 + 1];
        } else if (d < 288) {                      // user stats proj (16)
            const int j = d - 272;
            const float s0 = log1pf(user_stats[(long long)b * 2]);
            const float s1 = user_stats[(long long)b * 2 + 1];
            v = user_stat_b[j] + user_stat_w[j * 2] * s0 + user_stat_w[j * 2 + 1] * s1;
        } else if (d < 320) {                      // last category emb (32)
            v = category_emb[(long long)last_cat[b] * 32 + (d - 288)];
        } else {                                   // content projection (250 -> 64)
            const int o = d - 320;
            float acc = content_b[o];
            const float* wr = content_w + (long long)o * 250;
            for (int i = 0; i < 250; ++i) acc += s_lc[i] * wr[i];
            v = acc;
        }
        x[(long long)b * DIN + d] = v;
    }
}

// ---------------------------------------------------------------------------
// Kernel 2: double-tiled GEMM  out = epilogue( X_cat @ W^T + bias )
// Block = 256 threads = 8 wave32 waves. Output tile 64M x 128N, K tiled by 32.
// X (with optional concat X2) and W both staged in LDS (padded stride 33 to
// avoid 64-bank conflicts). Each wave owns one 16N column strip and 4 M
// sub-tiles -> each B fragment is reused by 4 v_wmma_f32_16x16x4_f32,
// cutting L2 weight traffic 4x vs the 16M version.
// MODE: 0 = bias, 1 = bias+relu, 2 = DCN cross: out = X + X0 * (z + bias)
// ---------------------------------------------------------------------------
template<int MODE>
__global__ __launch_bounds__(256) void gemm_wmma(
    const float* __restrict__ X, const float* __restrict__ X2,
    int in1, int in_total,
    const float* __restrict__ W, const float* __restrict__ bias,
    const float* __restrict__ X0, float* __restrict__ out, int N)
{
    __shared__ float xs [64  * XS_STRIDE];
    __shared__ float wsh[128 * WS_STRIDE];

    const int tid  = threadIdx.x;
    const int wave = tid >> 5;
    const int lane = tid & 31;
    const int m0   = blockIdx.x * 64;
    const int nb   = blockIdx.y * 128;
    const int n0   = nb + wave * 16;
    const int in2  = in_total - in1;

    const int m = lane & 15;     // A row / B col / D col within tile
    const int g = lane >> 4;     // lane group: selects K pair / M half

    v8f acc[4];
#pragma unroll
    for (int i = 0; i < 4; ++i)
        acc[i] = (v8f){0.f, 0.f, 0.f, 0.f, 0.f, 0.f, 0.f, 0.f};

    const int rr = tid >> 3;          // 0..31
    const int cc = (tid & 7) * 4;     // 0,4,...,28

    for (int kt0 = 0; kt0 < in_total; kt0 += KT) {
        __syncthreads();
        // stage X tile: 64 rows x 32 cols (float4, concat-aware; 384 % 32 == 0
        // so a 4-wide chunk never straddles the concat boundary)
#pragma unroll
        for (int it = 0; it < 2; ++it) {
            const int r = rr + it * 32;
            const long long row = m0 + r;
            const int kc = kt0 + cc;
            float4 v = (kc < in1)
                ? *(const float4*)(X  + row * in1 + kc)
                : *(const float4*)(X2 + row * in2 + (kc - in1));
            float* dst = xs + r * XS_STRIDE + cc;
            dst[0] = v.x; dst[1] = v.y; dst[2] = v.z; dst[3] = v.w;
        }
        // stage W tile: 128 rows x 32 cols
#pragma unroll
        for (int it = 0; it < 4; ++it) {
            const int r = rr + it * 32;
            const int wrow = nb + r;
            float4 v;
            if (wrow < N) v = *(const float4*)(W + (long long)wrow * in_total + kt0 + cc);
            else          v = (float4){0.f, 0.f, 0.f, 0.f};
            float* dst = wsh + r * WS_STRIDE + cc;
            dst[0] = v.x; dst[1] = v.y; dst[2] = v.z; dst[3] = v.w;
        }
        __syncthreads();

        if (n0 < N) {   // wave-uniform; EXEC stays all-ones inside
            const float* xrow = xs  + m * XS_STRIDE + 2 * g;
            const float* wrow = wsh + (wave * 16 + m) * WS_STRIDE + 2 * g;
#pragma unroll
            for (int ks = 0; ks < KT; ks += 4) {
                v2f bb; bb.x = wrow[ks]; bb.y = wrow[ks + 1];
#pragma unroll
                for (int ms = 0; ms < 4; ++ms) {
                    const float* ar = xrow + ms * 16 * XS_STRIDE;
                    v2f a; a.x = ar[ks]; a.y = ar[ks + 1];
                    acc[ms] = __builtin_amdgcn_wmma_f32_16x16x4_f32(
                        false, a, false, bb, (short)0, acc[ms], false, false);
                }
            }
        }
    }

    if (n0 < N) {
        const float bn = bias[n0 + m];
#pragma unroll
        for (int ms = 0; ms < 4; ++ms) {
#pragma unroll
            for (int v = 0; v < 8; ++v) {
                const int row = m0 + ms * 16 + v + 8 * g;  // D: M = vgpr + 8*laneGroup
                const int col = n0 + m;                    // D: N = lane[3:0]
                float val = acc[ms][v] + bn;
                if (MODE == 1) val = fmaxf(val, 0.f);
                if (MODE == 2) val = X[(long long)row * in_total + col]
                                   + X0[(long long)row * in_total + col] * val;
                out[(long long)row * N + col] = val;
            }
        }
    }
}

// ---------------------------------------------------------------------------
// Kernel 3: row-wise L2 normalize (one wave32 per 64-wide row)
// ---------------------------------------------------------------------------
__global__ void normalize_rows(const float* __restrict__ y, float* __restrict__ out)
{
    const int wave = threadIdx.x >> 5;
    const int lane = threadIdx.x & 31;
    const long long r = (long long)blockIdx.x * 8 + wave;
    const float* yr = y + r * 64;
    const float v0 = yr[lane];
    const float v1 = yr[lane + 32];
    float s = v0 * v0 + v1 * v1;
    for (int off = 16; off > 0; off >>= 1) s += __shfl_xor(s, off, 32);
    const float d = fmaxf(sqrtf(s), 1e-12f);
    out[r * 64 + lane]      = v0 / d;
    out[r * 64 + lane + 32] = v1 / d;
}

// ---------------------------------------------------------------------------
extern "C" void kernel_launch(void* const* d_in, const int* in_sizes, int n_in,
                              void* d_out, int out_size, void* d_ws, size_t ws_size,
                              hipStream_t stream) {
    (void)in_sizes; (void)n_in; (void)out_size; (void)ws_size;

    // setup_inputs() dict order, params dict flattened in insertion order.
    const float* hist_num_feats = (const float*)d_in[0];
    const float* user_stats     = (const float*)d_in[1];
    const float* last_content   = (const float*)d_in[2];
    const float* user_id_emb    = (const float*)d_in[3];
    const float* item_id_emb    = (const float*)d_in[4];
    const float* env_emb        = (const float*)d_in[5];
    const float* device_emb     = (const float*)d_in[6];
    const float* os_emb         = (const float*)d_in[7];
    const float* country_emb    = (const float*)d_in[8];
    const float* region_emb     = (const float*)d_in[9];
    const float* referrer_emb   = (const float*)d_in[10];
    const float* month_emb      = (const float*)d_in[11];
    const float* day_emb        = (const float*)d_in[12];
    const float* hour_emb       = (const float*)d_in[13];
    const float* minute_emb     = (const float*)d_in[14];
    const float* category_emb   = (const float*)d_in[15];
    const float* hist_num_w     = (const float*)d_in[16];
    const float* hist_num_b     = (const float*)d_in[17];
    const float* user_stat_w    = (const float*)d_in[18];
    const float* user_stat_b    = (const float*)d_in[19];
    const float* content_w      = (const float*)d_in[20];
    const float* content_b      = (const float*)d_in[21];
    const float* cross0_w       = (const float*)d_in[22];
    const float* cross0_b       = (const float*)d_in[23];
    const float* cross1_w       = (const float*)d_in[24];
    const float* cross1_b       = (const float*)d_in[25];
    const float* deep0_w        = (const float*)d_in[26];
    const float* deep0_b        = (const float*)d_in[27];
    const float* deep1_w        = (const float*)d_in[28];
    const float* deep1_b        = (const float*)d_in[29];
    const float* post_w         = (const float*)d_in[30];
    const float* post_b         = (const float*)d_in[31];
    const int* user_id      = (const int*)d_in[32];
    const int* env          = (const int*)d_in[33];
    const int* device_group = (const int*)d_in[34];
    const int* os_          = (const int*)d_in[35];
    const int* country      = (const int*)d_in[36];
    const int* region       = (const int*)d_in[37];
    const int* referrer     = (const int*)d_in[38];
    const int* month        = (const int*)d_in[39];
    const int* day          = (const int*)d_in[40];
    const int* hour         = (const int*)d_in[41];
    const int* minute       = (const int*)d_in[42];
    const int* history_ids  = (const int*)d_in[43];
    const int* last_cat     = (const int*)d_in[44];

    // workspace layout (f32)
    float* x   = (float*)d_ws;
    float* xc1 = x   + (size_t)B_ROWS * 384;
    float* xc2 = xc1 + (size_t)B_ROWS * 384;
    float* h0  = xc2 + (size_t)B_ROWS * 384;
    float* h1  = h0  + (size_t)B_ROWS * 256;
    float* y   = h1  + (size_t)B_ROWS * 128;

    build_features<<<B_ROWS, 128, 0, stream>>>(
        hist_num_feats, user_stats, last_content,
        user_id_emb, item_id_emb, env_emb, device_emb, os_emb, country_emb,
        region_emb, referrer_emb, month_emb, day_emb, hour_emb, minute_emb,
        category_emb, hist_num_w, hist_num_b, user_stat_w, user_stat_b,
        content_w, content_b,
        user_id, env, device_group, os_, country, region, referrer,
        month, day, hour, minute, history_ids, last_cat, x);

    const int MT = B_ROWS / 64;   // 256 row blocks of 64
    // cross0: xc1 = x + x * (x @ cross0_w^T + b)
    gemm_wmma<2><<<dim3(MT, 3), 256, 0, stream>>>(
        x, x, 384, 384, cross0_w, cross0_b, x, xc1, 384);
    // cross1: xc2 = xc1 + x * (xc1 @ cross1_w^T + b)
    gemm_wmma<2><<<dim3(MT, 3), 256, 0, stream>>>(
        xc1, xc1, 384, 384, cross1_w, cross1_b, x, xc2, 384);
    // deep0: h0 = relu(x @ deep0_w^T + b)
    gemm_wmma<1><<<dim3(MT, 2), 256, 0, stream>>>(
        x, x, 384, 384, deep0_w, deep0_b, nullptr, h0, 256);
    // deep1: h1 = relu(h0 @ deep1_w^T + b)
    gemm_wmma<1><<<dim3(MT, 1), 256, 0, stream>>>(
        h0, h0, 256, 256, deep1_w, deep1_b, nullptr, h1, 128);
    // post: y = concat(xc2, h1) @ post_w^T + b
    gemm_wmma<0><<<dim3(MT, 1), 256, 0, stream>>>(
        xc2, h1, 384, 512, post_w, post_b, nullptr, y, 64);

    normalize_rows<<<B_ROWS / 8, 256, 0, stream>>>(y, (float*)d_out);
}